// REANN_74921409511533
// MI455X (gfx1250) — compile-verified
//
#include <hip/hip_runtime.h>
#include <hip/hip_bf16.h>
#include <math.h>

// Problem constants (from reference)
#define NA_   10000   // atoms per image
#define NS_   4       // species
#define NMAX_ 32      // radial basis (== wave32 lanes, lane-parallel)
#define NO_   4       // orbital channels [1, x, y, z]
#define B_    4       // images
#define NM_   150000  // pairs per image
#define H1_   4       // MLP hidden 1
#define H2_   38      // MLP hidden 2

typedef __attribute__((ext_vector_type(2))) float v2f;
typedef __attribute__((ext_vector_type(8))) float v8f;

// ---------------------------------------------------------------------------
// Csn0[a][n] = species_params[sorted_numbers[a]][n]
// ---------------------------------------------------------------------------
__global__ void k_csn0(const float* __restrict__ sp, const int* __restrict__ zn,
                       float* __restrict__ csn0) {
    int i = blockIdx.x * blockDim.x + threadIdx.x;   // a*32 + n
    if (i < NA_ * NMAX_) {
        int a = i >> 5, n = i & 31;
        csn0[i] = sp[zn[a] * NMAX_ + n];
    }
}

// ---------------------------------------------------------------------------
// Pair accumulation / segment-sum, deterministic (no atomics):
// iidx is SORTED per image, so each (image, atom) owns a contiguous pair
// range found by binary search. One wave per atom; lane = radial channel n.
// bnl[b][a][l][n] = sum_{m: iidx[m]==a} fcut*angle_l*Csn[jidx[m]][n]*radial[n]
// ---------------------------------------------------------------------------
__global__ __launch_bounds__(32) void k_pairs(
    const float* __restrict__ disp, const int* __restrict__ iidx,
    const int* __restrict__ jidx, const int* __restrict__ zn,
    const float* __restrict__ alpha, const float* __restrict__ rs,
    const float* __restrict__ csn, int csn_img_stride,  // 0 => shared Csn0
    float* __restrict__ bnl) {
    const int b = blockIdx.x / NA_;
    const int a = blockIdx.x % NA_;
    const int lane = threadIdx.x;

    const int*   ib     = iidx + (size_t)b * NM_;
    const int*   jb     = jidx + (size_t)b * NM_;
    const float* db     = disp + (size_t)b * NM_ * 3;
    const float* csn_b  = csn  + (size_t)b * csn_img_stride;

    // lower_bound(a) and upper_bound(a) in sorted iidx  (uniform across wave)
    int lo = 0, hi = NM_;
    while (lo < hi) { int mid = (lo + hi) >> 1; if (ib[mid] <  a) lo = mid + 1; else hi = mid; }
    const int start = lo;
    hi = NM_;
    while (lo < hi) { int mid = (lo + hi) >> 1; if (ib[mid] <= a) lo = mid + 1; else hi = mid; }
    const int end = lo;

    const float PI_RC = 3.14159265358979323846f / 6.0f;  // pi / RCUT
    float acc0 = 0.f, acc1 = 0.f, acc2 = 0.f, acc3 = 0.f;

    for (int m = start; m < end; ++m) {
        // per-pair scalars (wave-uniform -> SALU/SMEM)
        float dx = db[3 * m + 0], dy = db[3 * m + 1], dz = db[3 * m + 2];
        int   jj = jb[m];
        int   ja = zn[jj];
        float dist = sqrtf(dx * dx + dy * dy + dz * dz);
        float ct   = cosf(dist * PI_RC) + 1.0f;
        float fcut = 0.25f * ct * ct;
        // per-lane (channel n) radial basis + gathered species coefficient
        float al = alpha[ja * NMAX_ + lane];
        float r0 = rs   [ja * NMAX_ + lane];
        float dd = dist - r0;
        float rad = expf(al * dd * dd);
        float cj  = csn_b[(size_t)jj * NMAX_ + lane];
        float v   = fcut * cj * rad;
        acc0 += v;                      // angle l=0 : 1
        acc1 += dx * v;                 // angle l=1 : x
        acc2 += dy * v;                 // angle l=2 : y
        acc3 += dz * v;                 // angle l=3 : z
    }

    float* p = bnl + ((((size_t)b * NA_ + a) * NO_) * NMAX_) + lane;
    p[0 * NMAX_] = acc0;
    p[1 * NMAX_] = acc1;
    p[2 * NMAX_] = acc2;
    p[3 * NMAX_] = acc3;
}

// ---------------------------------------------------------------------------
// Density contraction with f32 WMMA (numerically FMA-exact, matches f32 ref):
//   t[a,l,o] = sum_n W[l][n][o] * bnl[a,l,n]   (M=16 atoms, K=32, N=4 of 16)
//   rho[a,o] = sum_l t[a,l,o]^2
//
// One wave per 16-atom tile; 4 (l) x 8 (K/4) = 32 x V_WMMA_F32_16X16X4_F32.
//  - A tile (16 atoms x 4 l x 32 n = 8KB, contiguous in bnl) is staged into
//    LDS with GLOBAL_LOAD_ASYNC_TO_LDS_B128 (ASYNCcnt), then read as v2f
//    fragments via ds_load_b64.
//  - B tile (W, zero-padded to 16 columns) is staged into LDS with uniform,
//    unconditional stores — avoids divergent exec-masked global loads.
// Fragment layouts (ISA 7.12.2, 32-bit):
//  A: lane holds A[M=lane&15][K = 2*(lane>>4) + v]   (v = VGPR 0/1)
//  B: lane holds B[K = 2*(lane>>4) + v][N = lane&15]
//  D: VGPR r, lane -> t[M = r + 8*(lane>>4)][N = lane&15]
// ---------------------------------------------------------------------------
__global__ __launch_bounds__(32) void k_density(
    const float* __restrict__ bnl,
    const float* __restrict__ orb,   // orbital_params[pass], shape [2][NMAX][NO]
    float* __restrict__ rho) {
    const int tiles = NA_ / 16;                 // 625 (exact)
    const int b    = blockIdx.x / tiles;
    const int a0   = (blockIdx.x % tiles) * 16;
    const int lane = threadIdx.x;
    const int M    = lane & 15;
    const int half = lane >> 4;

    __shared__ __align__(16) float ldsA[16 * NO_ * NMAX_];   // 8KB: [atom][l][n]
    __shared__ float ldsW[2][NMAX_][16];                     // 4KB: zero-padded W

    // ---- async-stage A tile: 8KB contiguous slab of bnl -> LDS -------------
    const float* gA = bnl + (((size_t)b * NA_ + a0) * NO_) * NMAX_;
    {
        unsigned lbase = (unsigned)(uintptr_t)(&ldsA[0]);  // flat->LDS: low 32 bits
        #pragma unroll
        for (int t = 0; t < 16; ++t) {
            unsigned boff = (unsigned)((t * 32 + lane) * 16);  // 16B per lane
            unsigned laddr = lbase + boff;
            asm volatile("global_load_async_to_lds_b128 %0, %1, %2"
                         :: "v"(laddr), "v"(boff), "s"(gA)
                         : "memory");
        }
    }

    // ---- stage zero-padded W into LDS (uniform, unconditional) -------------
    #pragma unroll
    for (int t = 0; t < (2 * NMAX_ * 16) / 32; ++t)          // zero 1024 floats
        (&ldsW[0][0][0])[t * 32 + lane] = 0.f;
    #pragma unroll
    for (int sel = 0; sel < 2; ++sel)
        #pragma unroll
        for (int t = 0; t < (NMAX_ * NO_) / 32; ++t) {       // fill 128 per sel
            int e = t * 32 + lane;                           // e = n*NO + o
            ldsW[sel][e >> 2][e & 3] = orb[sel * (NMAX_ * NO_) + e];
        }

    asm volatile("s_wait_asynccnt 0x0" ::: "memory");        // A tile resident
    __syncthreads();

    float racc[8] = {0.f, 0.f, 0.f, 0.f, 0.f, 0.f, 0.f, 0.f};

    #pragma unroll
    for (int l = 0; l < NO_; ++l) {
        const int sel = (l == 0) ? 0 : 1;                    // OIDX = [0,1,1,1]
        const float* arow = &ldsA[(M * NO_ + l) * NMAX_];
        v8f c = {};
        #pragma unroll
        for (int k0 = 0; k0 < NMAX_; k0 += 4) {
            const int k = k0 + 2 * half;
            v2f av = *(const v2f*)&arow[k];                  // ds_load_b64
            v2f bv;
            bv.x = ldsW[sel][k + 0][M];
            bv.y = ldsW[sel][k + 1][M];
            // D = A(16x4 f32) * B(4x16 f32) + C   -> v_wmma_f32_16x16x4_f32
            c = __builtin_amdgcn_wmma_f32_16x16x4_f32(
                    false, av, false, bv, (short)0, c, false, false);
        }
        #pragma unroll
        for (int r = 0; r < 8; ++r) racc[r] += c[r] * c[r];  // sum over l of t^2
    }

    if (M < NO_) {
        #pragma unroll
        for (int r = 0; r < 8; ++r)
            rho[((size_t)b * NA_ + a0 + r + 8 * half) * NO_ + M] = racc[r];
    }
}

// ---------------------------------------------------------------------------
// Per-atom MLP update:  Csn1 = Csn0 + W3·tanh(W2·tanh(W1·rho + b1) + b2) + b3
// ---------------------------------------------------------------------------
__global__ void k_mlp(const float* __restrict__ rho, const float* __restrict__ csn0,
                      const float* __restrict__ w1, const float* __restrict__ b1,
                      const float* __restrict__ w2, const float* __restrict__ b2,
                      const float* __restrict__ w3, const float* __restrict__ b3,
                      float* __restrict__ csn1) {
    int i = blockIdx.x * blockDim.x + threadIdx.x;  // b*NA + a
    if (i >= B_ * NA_) return;
    float r[NO_];
    #pragma unroll
    for (int o = 0; o < NO_; ++o) r[o] = rho[(size_t)i * NO_ + o];

    float h1[H1_];
    #pragma unroll
    for (int j = 0; j < H1_; ++j) {
        float s = b1[j];
        #pragma unroll
        for (int o = 0; o < NO_; ++o) s += r[o] * w1[o * H1_ + j];
        h1[j] = tanhf(s);
    }
    float h2[H2_];
    for (int j = 0; j < H2_; ++j) {
        float s = b2[j];
        #pragma unroll
        for (int k = 0; k < H1_; ++k) s += h1[k] * w2[k * H2_ + j];
        h2[j] = tanhf(s);
    }
    const int a = i % NA_;
    for (int n = 0; n < NMAX_; ++n) {
        float s = b3[n];
        for (int k = 0; k < H2_; ++k) s += h2[k] * w3[k * NMAX_ + n];
        csn1[(size_t)i * NMAX_ + n] = csn0[a * NMAX_ + n] + s;
    }
}

// ---------------------------------------------------------------------------
extern "C" void kernel_launch(void* const* d_in, const int* in_sizes, int n_in,
                              void* d_out, int out_size, void* d_ws, size_t ws_size,
                              hipStream_t stream) {
    (void)in_sizes; (void)n_in; (void)out_size; (void)ws_size;
    const float* disp  = (const float*)d_in[0];
    const float* alpha = (const float*)d_in[1];
    const float* rs    = (const float*)d_in[2];
    const float* sp    = (const float*)d_in[3];
    const float* orb   = (const float*)d_in[4];   // [2][2][NMAX][NO]
    const float* w1    = (const float*)d_in[5];
    const float* b1    = (const float*)d_in[6];
    const float* w2    = (const float*)d_in[7];
    const float* b2    = (const float*)d_in[8];
    const float* w3    = (const float*)d_in[9];
    const float* b3    = (const float*)d_in[10];
    const int*   iidx  = (const int*)d_in[11];
    const int*   jidx  = (const int*)d_in[12];
    const int*   zn    = (const int*)d_in[13];
    float* out = (float*)d_out;

    // workspace layout (floats): csn0 | csn1 | bnl | rho1   (~27.5 MB)
    float* ws   = (float*)d_ws;
    float* csn0 = ws;                                          // NA*32
    float* csn1 = csn0 + (size_t)NA_ * NMAX_;                  // B*NA*32
    float* bnl  = csn1 + (size_t)B_ * NA_ * NMAX_;             // B*NA*4*32
    float* rho1 = bnl  + (size_t)B_ * NA_ * NO_ * NMAX_;       // B*NA*4

    const int tiles = NA_ / 16;

    k_csn0<<<(NA_ * NMAX_ + 255) / 256, 256, 0, stream>>>(sp, zn, csn0);

    // ---- pass 1 (shared Csn0, orbital_params[0]) ----
    k_pairs<<<B_ * NA_, 32, 0, stream>>>(disp, iidx, jidx, zn, alpha, rs,
                                         csn0, 0, bnl);
    k_density<<<B_ * tiles, 32, 0, stream>>>(bnl, orb, rho1);

    // ---- MLP update of Csn ----
    k_mlp<<<(B_ * NA_ + 127) / 128, 128, 0, stream>>>(rho1, csn0,
                                                      w1, b1, w2, b2, w3, b3, csn1);

    // ---- pass 2 (per-image Csn1, orbital_params[1]) ----
    k_pairs<<<B_ * NA_, 32, 0, stream>>>(disp, iidx, jidx, zn, alpha, rs,
                                         csn1, NA_ * NMAX_, bnl);
    k_density<<<B_ * tiles, 32, 0, stream>>>(bnl, orb + 2 * NMAX_ * NO_, out);
}